// GraphConvNetwork_2388001816782
// MI455X (gfx1250) — compile-verified
//
#include <hip/hip_runtime.h>

typedef __attribute__((ext_vector_type(2))) float v2f;
typedef __attribute__((ext_vector_type(8))) float v8f;

constexpr int IN_DIM = 128;
constexpr int HID    = 8;

// ---------------- utility kernels ----------------

__global__ void zero_f32(float* __restrict__ p, int n) {
  int i = blockIdx.x * blockDim.x + threadIdx.x;
  if (i < n) p[i] = 0.0f;
}

__global__ void degree_kernel(const int* __restrict__ src, const int* __restrict__ dst,
                              float* __restrict__ deg_s, float* __restrict__ deg_d, int nE) {
  int e = blockIdx.x * blockDim.x + threadIdx.x;
  if (e < nE) {
    atomicAdd(&deg_s[src[e]], 1.0f);
    atomicAdd(&deg_d[dst[e]], 1.0f);
  }
}

__global__ void norm_kernel(float* __restrict__ deg_s, float* __restrict__ deg_d, int n) {
  int i = blockIdx.x * blockDim.x + threadIdx.x;
  if (i < n) {
    deg_s[i] = rsqrtf(fmaxf(deg_s[i], 1.0f));
    deg_d[i] = rsqrtf(fmaxf(deg_d[i], 1.0f));
  }
}

// ---------------- layer 1 GEMM: h = (x * out_norm[:,None]) @ W1 ----------------
// One wave computes a 16-row (node) x 16-col tile with V_WMMA_F32_16X16X4_F32,
// K = 128 in steps of 4 (32 WMMAs). W1 is staged in LDS padded to 16 cols of
// zeros so no predication is needed around the WMMA (EXEC all-ones).
// The per-row out_norm scale is linear in A's rows, so it is applied to the
// D matrix at store time (8 muls/lane) instead of to every A fragment
// (64 muls/lane), letting global_load_b64 feed v_wmma directly.

__global__ __launch_bounds__(256) void gemm1_wmma(const float* __restrict__ x,
                                                  const float* __restrict__ W1,
                                                  const float* __restrict__ out_norm,
                                                  float* __restrict__ h, int nNodes) {
  __shared__ float w[IN_DIM * 16];  // 128 x 16, cols 8..15 = 0
  int tid = threadIdx.x;
  for (int i = tid; i < IN_DIM * 16; i += 256) {
    int r = i >> 4, cidx = i & 15;
    w[i] = (cidx < HID) ? W1[r * HID + cidx] : 0.0f;
  }
  __syncthreads();

  int wid  = tid >> 5;
  int lane = tid & 31;
  int m    = lane & 15;   // row within tile (A) / col N (B, D)
  int kh   = lane >> 4;   // K-half select

  int row0 = (blockIdx.x * 8 + wid) * 16;
  int node = row0 + m;
  int nodeClamped = node < nNodes ? node : nNodes - 1;  // clamped -> real data, no NaNs
  const float* xrow = x + (long)nodeClamped * IN_DIM;

  v8f acc = {};
#pragma unroll
  for (int kk = 0; kk < IN_DIM; kk += 4) {
    // A fragment (16x4 f32): lanes 0-15 -> K = kk,kk+1 ; lanes 16-31 -> K = kk+2,kk+3
    int k0 = kk + 2 * kh;
    v2f a;
    a.x = xrow[k0];
    a.y = xrow[k0 + 1];
    // B fragment (4x16 f32): lane gives N = m; VGPR0 -> K=k0, VGPR1 -> K=k0+1
    v2f b;
    b.x = w[k0 * 16 + m];
    b.y = w[(k0 + 1) * 16 + m];
    acc = __builtin_amdgcn_wmma_f32_16x16x4_f32(
        /*neg_a=*/false, a, /*neg_b=*/false, b,
        /*c_mod=*/(short)0, acc, /*reuse_a=*/false, /*reuse_b=*/false);
  }

  // D layout: VGPR r -> row M = r + 8*kh, col N = m. Apply out_norm per D row.
  if (m < HID) {
#pragma unroll
    for (int r = 0; r < 8; ++r) {
      int nr = row0 + r + 8 * kh;
      if (nr < nNodes) h[(long)nr * HID + m] = acc[r] * out_norm[nr];
    }
  }
}

// ---------------- SpMM 1: agg[dst] += h[src] (8 floats/edge) ----------------

__global__ void spmm1_kernel(const int* __restrict__ src, const int* __restrict__ dst,
                             const float* __restrict__ h, float* __restrict__ agg, int nE) {
  int e = blockIdx.x * blockDim.x + threadIdx.x;
  if (e >= nE) return;
  int s = src[e], d = dst[e];
  const float4* hs = (const float4*)(h + (long)s * HID);
  float4 h0 = hs[0];
  float4 h1 = hs[1];
  float* ad = agg + (long)d * HID;
  atomicAdd(&ad[0], h0.x); atomicAdd(&ad[1], h0.y);
  atomicAdd(&ad[2], h0.z); atomicAdd(&ad[3], h0.w);
  atomicAdd(&ad[4], h1.x); atomicAdd(&ad[5], h1.y);
  atomicAdd(&ad[6], h1.z); atomicAdd(&ad[7], h1.w);
}

// ------------- fused: z = relu(agg*in_norm + b1); g2 = out_norm * (z @ W2) -------------

__global__ void layer2_node(const float* __restrict__ agg, const float* __restrict__ in_norm,
                            const float* __restrict__ out_norm, const float* __restrict__ b1,
                            const float* __restrict__ W2, float* __restrict__ g2, int n) {
  int v = blockIdx.x * blockDim.x + threadIdx.x;
  if (v >= n) return;
  float innv = in_norm[v];
  const float* av = agg + (long)v * HID;
  float s = 0.0f;
#pragma unroll
  for (int j = 0; j < HID; ++j) {
    float z = fmaf(av[j], innv, b1[j]);
    z = z > 0.0f ? z : 0.0f;
    s = fmaf(z, W2[j], s);
  }
  g2[v] = s * out_norm[v];
}

// ---------------- SpMM 2: outacc[dst] += g2[src] ----------------

__global__ void spmm2_kernel(const int* __restrict__ src, const int* __restrict__ dst,
                             const float* __restrict__ g2, float* __restrict__ outacc, int nE) {
  int e = blockIdx.x * blockDim.x + threadIdx.x;
  if (e < nE) atomicAdd(&outacc[dst[e]], g2[src[e]]);
}

__global__ void final_kernel(const float* __restrict__ outacc, const float* __restrict__ in_norm,
                             const float* __restrict__ b2, float* __restrict__ out, int n) {
  int v = blockIdx.x * blockDim.x + threadIdx.x;
  if (v < n) out[v] = fmaf(outacc[v], in_norm[v], b2[0]);
}

// ---------------- launch ----------------

extern "C" void kernel_launch(void* const* d_in, const int* in_sizes, int n_in,
                              void* d_out, int out_size, void* d_ws, size_t ws_size,
                              hipStream_t stream) {
  const float* x  = (const float*)d_in[0];
  const float* W1 = (const float*)d_in[1];
  const float* b1 = (const float*)d_in[2];
  const float* W2 = (const float*)d_in[3];
  const float* b2 = (const float*)d_in[4];
  const int*  src = (const int*)d_in[5];
  const int*  dst = (const int*)d_in[6];

  const int nN = in_sizes[0] / IN_DIM;   // 500000
  const int nE = in_sizes[5];            // 16000000

  float* ws     = (float*)d_ws;
  float* deg_s  = ws;                    // N  -> becomes out_norm
  float* deg_d  = ws + (long)nN;         // N  -> becomes in_norm
  float* h      = ws + 2L * nN;          // 8N
  float* agg    = ws + 10L * nN;         // 8N
  float* g2     = ws + 18L * nN;         // N
  float* outacc = ws + 19L * nN;         // N
  float* out    = (float*)d_out;

  const int TB = 256;
  dim3 blk(TB);

  zero_f32<<<(2 * nN + TB - 1) / TB, blk, 0, stream>>>(deg_s, 2 * nN);  // degs contiguous
  zero_f32<<<(8 * nN + TB - 1) / TB, blk, 0, stream>>>(agg, 8 * nN);
  zero_f32<<<(nN + TB - 1) / TB, blk, 0, stream>>>(outacc, nN);

  degree_kernel<<<(nE + TB - 1) / TB, blk, 0, stream>>>(src, dst, deg_s, deg_d, nE);
  norm_kernel<<<(nN + TB - 1) / TB, blk, 0, stream>>>(deg_s, deg_d, nN);

  gemm1_wmma<<<(nN + 127) / 128, blk, 0, stream>>>(x, W1, deg_s, h, nN);

  spmm1_kernel<<<(nE + TB - 1) / TB, blk, 0, stream>>>(src, dst, h, agg, nE);
  layer2_node<<<(nN + TB - 1) / TB, blk, 0, stream>>>(agg, deg_d, deg_s, b1, W2, g2, nN);
  spmm2_kernel<<<(nE + TB - 1) / TB, blk, 0, stream>>>(src, dst, g2, outacc, nE);
  final_kernel<<<(nN + TB - 1) / TB, blk, 0, stream>>>(outacc, deg_d, b2, out, nN);
}